// RoPE3DEncoder_5377299055154
// MI455X (gfx1250) — compile-verified
//
#include <hip/hip_runtime.h>
#include <cstdint>

// 3D RoPE cos/sin table generator for MI455X (gfx1250).
// Output (201 MB) is pure-store-bound: ~8.6 us floor at 23.3 TB/s.
// Strategy: per-block LDS tables of the 5120 unique angles, then stream the
// output with GLOBAL_STORE_ASYNC_FROM_LDS_B128 (per-lane LDS gather -> global),
// tracked by ASYNCcnt. NT temporal hint: output is written once and never
// re-read by the kernel, so don't let it rinse through L2.

#define RP_DIM   192
#define AX_DIM   64     // DIM/3 for all three axes (192/3 = 64, DIM_T = 192-128 = 64)
#define HALF     32     // unique frequencies per axis
#define TAB_CAP  5120   // (64 + 64 + 32) * 32 max table entries
#define THREADS  256

__global__ __launch_bounds__(THREADS) void rope3d_async_kernel(
    const int* __restrict__ pH, const int* __restrict__ pW, const int* __restrict__ pF,
    float* __restrict__ out, unsigned total4)
{
    __shared__ alignas(16) float cosT[TAB_CAP];
    __shared__ alignas(16) float sinT[TAB_CAP];

    const unsigned H = (unsigned)pH[0];
    const unsigned W = (unsigned)pW[0];
    const unsigned F = (unsigned)pF[0];
    const unsigned yBase = W * HALF;
    const unsigned tBase = (W + H) * HALF;
    const unsigned ntab  = (W + H + F) * HALF;

    // ---- Build per-axis cos/sin angle tables (5120 sincos per block) ----
    for (unsigned i = threadIdx.x; i < ntab && i < TAB_CAP; i += THREADS) {
        unsigned rel = (i < yBase) ? i : ((i < tBase) ? (i - yBase) : (i - tBase));
        unsigned pos = rel >> 5;
        unsigned j   = rel & 31u;
        // inv_freq = 10000^(-2j/64) = exp(-j * ln(10000)/32)
        float inv = expf(-0.28782313662425572f * (float)j);
        float ang = (float)pos * inv;
        cosT[i] = cosf(ang);
        sinT[i] = sinf(ang);
    }
    __syncthreads();

    // ---- Stream output: each float4 of the result is a contiguous 16B slice
    //      of the LDS tables -> async store straight from LDS. ----
    const unsigned gsize = gridDim.x * THREADS;
    float4* outCos = reinterpret_cast<float4*>(out);
    float4* outSin = outCos + total4;

    for (unsigned c4 = blockIdx.x * THREADS + threadIdx.x; c4 < total4; c4 += gsize) {
        unsigned n = c4 / 48u;                 // row index (48 float4 per 192-wide row)
        unsigned d = (c4 - n * 48u) << 2;      // feature start within row
        unsigned x  = n % W;
        unsigned nw = n / W;
        unsigned y  = nw % H;
        unsigned t  = nw / H;

        unsigned pos, base, dd;
        if (d < AX_DIM)            { pos = x; base = 0u;    dd = d; }
        else if (d < 2u * AX_DIM)  { pos = y; base = yBase; dd = d - AX_DIM; }
        else                       { pos = t; base = tBase; dd = d - 2u * AX_DIM; }
        // concat([f, f]) duplication -> fold feature offset into [0,32)
        unsigned idx = base + (pos << 5) + (dd & 31u);

        unsigned ldsC = (unsigned)(uintptr_t)(&cosT[idx]);   // low 32 bits of flat
        unsigned ldsS = (unsigned)(uintptr_t)(&sinT[idx]);   // LDS-aperture addr
        unsigned long long gC = (unsigned long long)(uintptr_t)(outCos + c4);
        unsigned long long gS = (unsigned long long)(uintptr_t)(outSin + c4);

        asm volatile("global_store_async_from_lds_b128 %0, %1, off th:TH_STORE_NT"
                     :: "v"(gC), "v"(ldsC) : "memory");
        asm volatile("global_store_async_from_lds_b128 %0, %1, off th:TH_STORE_NT"
                     :: "v"(gS), "v"(ldsS) : "memory");
    }

    // Drain all async LDS->global transfers before wave exit.
    __builtin_amdgcn_s_wait_asynccnt(0);
}

extern "C" void kernel_launch(void* const* d_in, const int* in_sizes, int n_in,
                              void* d_out, int out_size, void* d_ws, size_t ws_size,
                              hipStream_t stream) {
    (void)in_sizes; (void)n_in; (void)d_ws; (void)ws_size;
    // setup_inputs order: x (unused), height, width, frames
    const int* pH = (const int*)d_in[1];
    const int* pW = (const int*)d_in[2];
    const int* pF = (const int*)d_in[3];
    float* out = (float*)d_out;

    unsigned total  = (unsigned)(out_size / 2);  // elements per output (cos or sin)
    unsigned total4 = total / 4u;                // 128-bit chunks per output
    unsigned need   = (total4 + THREADS - 1) / THREADS;
    unsigned blocks = 2048u;                      // persistent grid-stride
    if (blocks > need) blocks = need ? need : 1u;

    rope3d_async_kernel<<<dim3(blocks), dim3(THREADS), 0, stream>>>(
        pH, pW, pF, out, total4);
}